// LSTMAutoencoder_10230612099891
// MI455X (gfx1250) — compile-verified
//
#include <hip/hip_runtime.h>
#include <hip/hip_bf16.h>
#include <cstdint>

typedef __attribute__((ext_vector_type(2))) float v2f;
typedef __attribute__((ext_vector_type(8))) float v8f;

static constexpr int T_ = 256;

__device__ __forceinline__ v8f wmma_f32_16x16x4(v2f a, v2f b, v8f c) {
  // D = A(16x4, f32) * B(4x16, f32) + C(16x16, f32)
  return __builtin_amdgcn_wmma_f32_16x16x4_f32(false, a, false, b, (short)0, c,
                                               false, false);
}

// Fast saturating activations: v_exp_f32 + v_rcp_f32 only (no branches).
__device__ __forceinline__ float fast_sigmoid(float x) {
  return __builtin_amdgcn_rcpf(1.0f + __expf(-x));
}
__device__ __forceinline__ float fast_tanh(float x) {
  // tanh(x) = 1 - 2/(exp(2x)+1); exp overflow/underflow saturate to +-1.
  return fmaf(-2.0f, __builtin_amdgcn_rcpf(1.0f + __expf(2.0f * x)), 1.0f);
}

// One workgroup owns a 16-row batch tile for the whole sequence.
// Wave w owns hidden columns [w*16, w*16+16) of all four gates (i,f,g,o).
// Per step: gates(16 x 4H) = [x_t | h](16 x (I+H)) @ W^T + (bih+bhh), as
// chained V_WMMA_F32_16X16X4_F32. h and x_t are double-buffered in LDS so a
// single barrier per step suffices; x[t+1] is prefetched into registers
// during step t. Weight (B) fragments are register-resident.
// Intermediate sequences use (T,B,H) layout for contiguous per-step tiles;
// IN_BT marks the external (B,T,I) input layout (layer e1 only).
template <int I, int H, bool REPEAT, bool IN_BT, bool STORE_SEQ, bool STORE_LAST>
__global__ __launch_bounds__(128) void lstm_scan_kernel(
    const float* __restrict__ x,     // (B,T,I) | (T,B,I) | (B,I) latent if REPEAT
    const float* __restrict__ wih,   // (4H, I) row-major
    const float* __restrict__ whh,   // (4H, H) row-major
    const float* __restrict__ bih,   // (4H)
    const float* __restrict__ bhh,   // (4H)
    float* __restrict__ out_seq,     // (T,B,H) if STORE_SEQ
    float* __restrict__ out_last,    // (B,H)   if STORE_LAST
    int B)
{
  constexpr int XS = I + 2;              // padded strides: conflict-free, 8B-aligned
  constexpr int HS = H + 2;
  constexpr int NXB = REPEAT ? 1 : 2;
  __shared__ float xsh[NXB * 16 * XS];
  __shared__ float hsh[2 * 16 * HS];

  const int tid   = threadIdx.x;
  const int lane  = tid & 31;
  const int wave  = tid >> 5;            // 0 .. H/16-1
  const int c0    = wave * 16;           // hidden-column base for this wave
  const int n     = lane & 15;           // N (and A-row M) index for this lane
  const int khalf = (lane >> 4) * 2;     // A/B sub-k for upper/lower half-wave
  const int crow  = (lane >> 4) * 8;     // C/D row base for this lane
  const int b0    = blockIdx.x * 16;
  const int nthr  = 2 * H;               // == blockDim.x

  // Register-resident B fragments: B elem (k,col) = W^T with W=[wih|whh].
  v2f bx[4][I / 4];
  v2f bh[4][H / 4];
  float bias[4];
#pragma unroll
  for (int g = 0; g < 4; ++g) {
    const int col = g * H + c0 + n;
    const float* wi = wih + (size_t)col * I;
    const float* wh = whh + (size_t)col * H;
#pragma unroll
    for (int ks = 0; ks < I / 4; ++ks) {
      const int k = ks * 4 + khalf;
      v2f b; b.x = wi[k]; b.y = wi[k + 1];
      bx[g][ks] = b;
    }
#pragma unroll
    for (int ks = 0; ks < H / 4; ++ks) {
      const int k = ks * 4 + khalf;
      v2f b; b.x = wh[k]; b.y = wh[k + 1];
      bh[g][ks] = b;
    }
    bias[g] = bih[col] + bhh[col];
  }

  // Zero h buffer 0.
  for (int idx = tid; idx < 16 * H; idx += nthr) {
    const int r = idx / H, k = idx % H;
    hsh[r * HS + k] = 0.0f;
  }
  // Stage initial x tile (latent if REPEAT, else x[t=0]) into x buffer 0.
  for (int idx = tid; idx < 8 * I; idx += nthr) {
    const int fo = 2 * idx;
    const int r = fo / I, c = fo % I;
    const float* src =
        REPEAT ? &x[(size_t)(b0 + r) * I + c]
               : (IN_BT ? &x[(size_t)(b0 + r) * T_ * I + c]           // t=0
                        : &x[(size_t)(b0 + r) * I + c]);              // t=0, TBH
    *(v2f*)&xsh[r * XS + c] = *(const v2f*)src;
  }
  __syncthreads();

  // REPEAT: input projection is time-invariant -> fold into base accumulators.
  v8f base[4];
  if (REPEAT) {
#pragma unroll
    for (int g = 0; g < 4; ++g)
#pragma unroll
      for (int v = 0; v < 8; ++v) base[g][v] = bias[g];
#pragma unroll
    for (int ks = 0; ks < I / 4; ++ks) {
      const v2f a = *(const v2f*)&xsh[n * XS + ks * 4 + khalf];
#pragma unroll
      for (int g = 0; g < 4; ++g) base[g] = wmma_f32_16x16x4(a, bx[g][ks], base[g]);
    }
  }

  v8f cst;                               // cell state, C/D layout
#pragma unroll
  for (int v = 0; v < 8; ++v) cst[v] = 0.0f;

  constexpr int XPT = REPEAT ? 1 : (8 * I) / (2 * H);  // v2f chunks per thread
  int p = 0;

  for (int t = 0; t < T_; ++t) {
    // Prefetch x[t+1] into registers (off the critical path).
    v2f xr[XPT];
    if (!REPEAT) {
      if (t + 1 < T_) {
#pragma unroll
        for (int j = 0; j < XPT; ++j) {
          const int fo = 2 * (tid + j * nthr);
          const int r = fo / I, c = fo % I;
          const float* src =
              IN_BT ? &x[((size_t)(b0 + r) * T_ + (t + 1)) * I + c]
                    : &x[((size_t)(t + 1) * B + (b0 + r)) * I + c];
          xr[j] = *(const v2f*)src;
        }
      }
    }

    v8f acc[4];
    if (REPEAT) {
#pragma unroll
      for (int g = 0; g < 4; ++g) acc[g] = base[g];
    } else {
#pragma unroll
      for (int g = 0; g < 4; ++g)
#pragma unroll
        for (int v = 0; v < 8; ++v) acc[g][v] = bias[g];
    }

    if (!REPEAT) {
      const float* xb = &xsh[p * 16 * XS];
#pragma unroll
      for (int ks = 0; ks < I / 4; ++ks) {
        const v2f a = *(const v2f*)&xb[n * XS + ks * 4 + khalf];
#pragma unroll
        for (int g = 0; g < 4; ++g) acc[g] = wmma_f32_16x16x4(a, bx[g][ks], acc[g]);
      }
    }
    {
      const float* hb = &hsh[p * 16 * HS];
#pragma unroll
      for (int ks = 0; ks < H / 4; ++ks) {
        const v2f a = *(const v2f*)&hb[n * HS + ks * 4 + khalf];
#pragma unroll
        for (int g = 0; g < 4; ++g) acc[g] = wmma_f32_16x16x4(a, bh[g][ks], acc[g]);
      }
    }

    // Elementwise LSTM cell (torch gate order i,f,g,o).
    v8f hnew;
#pragma unroll
    for (int v = 0; v < 8; ++v) {
      const float ig = fast_sigmoid(acc[0][v]);
      const float fg = fast_sigmoid(acc[1][v]);
      const float gg = fast_tanh(acc[2][v]);
      const float og = fast_sigmoid(acc[3][v]);
      const float cv = fg * cst[v] + ig * gg;
      cst[v] = cv;
      hnew[v] = og * fast_tanh(cv);
    }

    // Commit h_new and prefetched x to the alternate buffers.
    {
      float* hd = &hsh[(p ^ 1) * 16 * HS];
#pragma unroll
      for (int v = 0; v < 8; ++v)
        hd[(crow + v) * HS + c0 + n] = hnew[v];
    }
    if (!REPEAT) {
      if (t + 1 < T_) {
        float* xd = &xsh[(p ^ 1) * 16 * XS];
#pragma unroll
        for (int j = 0; j < XPT; ++j) {
          const int fo = 2 * (tid + j * nthr);
          const int r = fo / I, c = fo % I;
          *(v2f*)&xd[r * XS + c] = xr[j];
        }
      }
    }

    if (STORE_SEQ) {
      // (T,B,H): contiguous 64B row-chunks per lane group.
#pragma unroll
      for (int v = 0; v < 8; ++v)
        out_seq[((size_t)t * B + (b0 + crow + v)) * H + c0 + n] = hnew[v];
    }
    if (STORE_LAST) {
      if (t == T_ - 1) {
#pragma unroll
        for (int v = 0; v < 8; ++v)
          out_last[(size_t)(b0 + crow + v) * H + c0 + n] = hnew[v];
      }
    }

    __syncthreads();                     // single barrier per step
    p ^= 1;
  }
}

// reconstruction = seq(T*B x 64) @ out_w^T(64 x 64) + out_b, written back in
// (B,T,64) order. 2D grid: blockIdx.x = batch tile, blockIdx.y = t (no integer
// division anywhere).
__global__ __launch_bounds__(128) void out_proj_kernel(
    const float* __restrict__ in,    // (T, B, 64)
    const float* __restrict__ w,     // (64, 64) row-major (o, h)
    const float* __restrict__ bias,  // (64)
    float* __restrict__ out,         // (B, T, 64)
    int B)
{
  __shared__ float at[16 * 66];
  const int tid   = threadIdx.x;
  const int lane  = tid & 31;
  const int wave  = tid >> 5;        // 4 waves -> 4 N-tiles of 16
  const int c0    = wave * 16;
  const int n     = lane & 15;
  const int khalf = (lane >> 4) * 2;
  const int crow  = (lane >> 4) * 8;
  const int bb    = blockIdx.x * 16;
  const int t     = blockIdx.y;
  const size_t m0 = (size_t)t * B + bb;  // row base in (T,B) flat order

  v2f bfrag[16];
  {
    const float* wr = w + (size_t)(c0 + n) * 64;
#pragma unroll
    for (int ks = 0; ks < 16; ++ks) {
      v2f b;
      b.x = wr[ks * 4 + khalf];
      b.y = wr[ks * 4 + khalf + 1];
      bfrag[ks] = b;
    }
  }

  for (int idx = tid; idx < 16 * 64; idx += blockDim.x) {
    const int r = idx >> 6, k = idx & 63;
    at[r * 66 + k] = in[(m0 + r) * 64 + k];
  }
  __syncthreads();

  const float bv = bias[c0 + n];
  v8f acc;
#pragma unroll
  for (int v = 0; v < 8; ++v) acc[v] = bv;

#pragma unroll
  for (int ks = 0; ks < 16; ++ks) {
    const v2f a = *(const v2f*)&at[n * 66 + ks * 4 + khalf];
    acc = wmma_f32_16x16x4(a, bfrag[ks], acc);
  }

#pragma unroll
  for (int v = 0; v < 8; ++v)
    out[((size_t)(bb + crow + v) * T_ + t) * 64 + c0 + n] = acc[v];
}

__global__ void copy_kernel(const float* __restrict__ src,
                            float* __restrict__ dst, int nelem) {
  const int i = blockIdx.x * 256 + threadIdx.x;
  if (i < nelem) dst[i] = src[i];
}

extern "C" void kernel_launch(void* const* d_in, const int* in_sizes, int n_in,
                              void* d_out, int out_size, void* d_ws, size_t ws_size,
                              hipStream_t stream) {
  (void)n_in; (void)out_size; (void)ws_size;
  const float* x = (const float*)d_in[0];
  const int Bv = in_sizes[0] / (T_ * 64);   // 2048

  // params flattened in dict order: e1,e2,e3,d1,d2,d3 (wih,whh,bih,bhh each),
  // then out_w, out_b.
  const float* const* p = (const float* const*)(d_in + 1);
  const float* e1_[4] = {p[0], p[1], p[2], p[3]};
  const float* e2_[4] = {p[4], p[5], p[6], p[7]};
  const float* e3_[4] = {p[8], p[9], p[10], p[11]};
  const float* d1_[4] = {p[12], p[13], p[14], p[15]};
  const float* d2_[4] = {p[16], p[17], p[18], p[19]};
  const float* d3_[4] = {p[20], p[21], p[22], p[23]};
  const float* out_w  = p[24];
  const float* out_b  = p[25];

  float* ws0    = (float*)d_ws;                         // T*B*64 floats
  float* wslat  = ws0 + (size_t)Bv * T_ * 64;           // B*16 floats
  float* rec    = (float*)d_out;                        // final reconstruction
  float* latout = rec + (size_t)Bv * T_ * 64;           // final latent
  float* scratch = rec;                                 // d_out doubles as ping buffer

  const dim3 grid(Bv / 16);

  // Encoder (intermediates in (T,B,H) layout)
  lstm_scan_kernel<64, 64, false, true, true, false><<<grid, 128, 0, stream>>>(
      x, e1_[0], e1_[1], e1_[2], e1_[3], ws0, nullptr, Bv);
  lstm_scan_kernel<64, 32, false, false, true, false><<<grid, 64, 0, stream>>>(
      ws0, e2_[0], e2_[1], e2_[2], e2_[3], scratch, nullptr, Bv);
  lstm_scan_kernel<32, 16, false, false, false, true><<<grid, 32, 0, stream>>>(
      scratch, e3_[0], e3_[1], e3_[2], e3_[3], nullptr, wslat, Bv);
  // Decoder (d1 consumes the latent; its x-projection is hoisted out of the scan)
  lstm_scan_kernel<16, 32, true, false, true, false><<<grid, 64, 0, stream>>>(
      wslat, d1_[0], d1_[1], d1_[2], d1_[3], ws0, nullptr, Bv);
  lstm_scan_kernel<32, 64, false, false, true, false><<<grid, 128, 0, stream>>>(
      ws0, d2_[0], d2_[1], d2_[2], d2_[3], scratch, nullptr, Bv);
  lstm_scan_kernel<64, 64, false, false, true, false><<<grid, 128, 0, stream>>>(
      scratch, d3_[0], d3_[1], d3_[2], d3_[3], ws0, nullptr, Bv);

  // Output projection (reorders (T,B)->(B,T)) + latent copy into d_out tail
  out_proj_kernel<<<dim3(Bv / 16, T_), 128, 0, stream>>>(ws0, out_w, out_b, rec, Bv);
  copy_kernel<<<dim3((Bv * 16 + 255) / 256), 256, 0, stream>>>(wslat, latout, Bv * 16);
}